// CurriculumActorCritic_16466904613265
// MI455X (gfx1250) — compile-verified
//
#include <hip/hip_runtime.h>
#include <hip/hip_bf16.h>

// ---------------------------------------------------------------------------
// Sizes (fixed by the reference)
// ---------------------------------------------------------------------------
#define BB 128
#define HH 64
#define WW 64
#define NC 16
#define HID 256
#define EE 32
#define DD 64
#define NHD 4
#define HDD 16
#define T 1024          // NC * HH distinct (color,row) table entries

typedef __attribute__((ext_vector_type(2))) float v2f;
typedef __attribute__((ext_vector_type(8))) float v8f;

// ---------------------------------------------------------------------------
// Kernel 1: build X table (1024 x 64): X[t] = [color_embed[t>>6], row_embed[t&63]]
// ---------------------------------------------------------------------------
__global__ void build_x_kernel(const float* __restrict__ color_embed,
                               const float* __restrict__ row_embed,
                               float* __restrict__ X) {
    int i = blockIdx.x * blockDim.x + threadIdx.x;   // 0 .. 65535
    int t = i >> 6;
    int e = i & 63;
    int color = t >> 6;
    int row   = t & 63;
    X[i] = (e < EE) ? color_embed[color * EE + e] : row_embed[row * EE + (e - EE)];
}

// ---------------------------------------------------------------------------
// Kernel 2: fp32 WMMA GEMM  C[M,N] = act(A[M,K] @ Wt[K,N] + bias[N])
// One wave32 computes one 16x16 tile via V_WMMA_F32_16X16X4_F32.
// A fragment (16x4, ISA 7.12.2): lane l<16 -> {A[m0+l][k0], A[m0+l][k0+1]},
//                                lane l>=16 -> {A[m0+l-16][k0+2], A[m0+l-16][k0+3]}
// B fragment (4x16) mirrored with N across lanes.
// C/D: VGPR j holds row m0+j (lanes 0-15) / m0+j+8 (lanes 16-31), col = n0+(l&15)
// ---------------------------------------------------------------------------
__global__ void wmma_gemm_f32_kernel(const float* __restrict__ A,
                                     const float* __restrict__ Wt,
                                     const float* __restrict__ bias,
                                     float* __restrict__ C,
                                     int M, int K, int N, int relu) {
    const int lane  = threadIdx.x & 31;
    const int wave  = threadIdx.x >> 5;
    const int waves = blockDim.x >> 5;
    const int tile  = blockIdx.x * waves + wave;
    const int tilesN = N >> 4;
    const int tm = (tile / tilesN) << 4;
    const int tn = (tile % tilesN) << 4;
    if (tm >= M) return;                 // uniform per-wave -> EXEC stays all-ones

    const int half = lane >> 4;          // 0 | 1
    const int l    = lane & 15;

    v8f acc = {};
    for (int k0 = 0; k0 < K; k0 += 4) {
        const int ka = k0 + (half << 1);
        v2f a, b;
        a.x = A[(tm + l) * K + ka];
        a.y = A[(tm + l) * K + ka + 1];
        b.x = Wt[ka * N + tn + l];
        b.y = Wt[(ka + 1) * N + tn + l];
        acc = __builtin_amdgcn_wmma_f32_16x16x4_f32(
            /*neg_a=*/false, a, /*neg_b=*/false, b,
            /*c_mod=*/(short)0, acc, /*reuse_a=*/false, /*reuse_b=*/false);
    }

    const int col = tn + l;
    const float bb = bias[col];
#pragma unroll
    for (int j = 0; j < 8; ++j) {
        float v = acc[j] + bb;
        if (relu) v = fmaxf(v, 0.0f);
        C[(tm + j + 8 * half) * N + col] = v;
    }
}

// ---------------------------------------------------------------------------
// Kernel 3: per-batch histograms over (color,row) + first masked pixel index
// ---------------------------------------------------------------------------
__global__ void hist_kernel(const int* __restrict__ obs,
                            int* __restrict__ hist_in,
                            int* __restrict__ hist_out,
                            int* __restrict__ first_idx) {
    const int b = blockIdx.x;
    __shared__ int hin[T];
    __shared__ int hout[T];
    __shared__ int hidx;
    for (int i = threadIdx.x; i < T; i += blockDim.x) { hin[i] = 0; hout[i] = 0; }
    if (threadIdx.x == 0) hidx = HH * WW;   // sentinel: no mask
    __syncthreads();

    const int* in   = obs + b * 3 * HH * WW;
    const int* outg = in + HH * WW;
    const int* msk  = in + 2 * HH * WW;
    for (int p = threadIdx.x; p < HH * WW; p += blockDim.x) {
        int row = p >> 6;
        atomicAdd(&hin[in[p] * HH + row], 1);
        atomicAdd(&hout[outg[p] * HH + row], 1);
        if (msk[p] != 0) atomicMin(&hidx, p);
    }
    __syncthreads();
    for (int i = threadIdx.x; i < T; i += blockDim.x) {
        hist_in[b * T + i]  = hin[i];
        hist_out[b * T + i] = hout[i];
    }
    if (threadIdx.x == 0) first_idx[b] = hidx;
}

// ---------------------------------------------------------------------------
// Kernel 4: per-batch epilogue (one block of 256 per batch):
//   cur, q, count-weighted softmax attention over 1024 unique keys,
//   output_context from hist_out x F_out table, head MLPs, write 17 outputs.
// ---------------------------------------------------------------------------
__global__ void finalize_kernel(const float* __restrict__ row_embed,
                                const float* __restrict__ col_embed,
                                const float* __restrict__ Wq, const float* __restrict__ bq,
                                const float* __restrict__ Wo, const float* __restrict__ bo,
                                const float* __restrict__ Ktab,
                                const float* __restrict__ Vtab,
                                const float* __restrict__ Fout,
                                const int* __restrict__ hist_in,
                                const int* __restrict__ hist_out,
                                const int* __restrict__ first_idx,
                                const float* __restrict__ p_W1, const float* __restrict__ p_b1,
                                const float* __restrict__ p_W2, const float* __restrict__ p_b2,
                                const float* __restrict__ v_W1, const float* __restrict__ v_b1,
                                const float* __restrict__ v_W2, const float* __restrict__ v_b2,
                                float* __restrict__ out) {
    const int b   = blockIdx.x;
    const int tid = threadIdx.x;

    __shared__ float cur[DD], q[DD], attended[DD], att_o[DD], ctx[DD];
    __shared__ float combined[3 * DD];
    __shared__ float h1p[HID], h1v[HID];
    __shared__ float red[256];
    __shared__ float headmax[NHD];
    __shared__ float headnum[NHD][HDD];
    __shared__ float headden[NHD];
    __shared__ float ctxpart[4][DD];

    // ---- cur = has ? [row_embed[i], col_embed[j]] : 0
    const int idx = first_idx[b];
    const bool has = idx < HH * WW;
    const int ri = (idx >> 6) & 63;
    const int cj = idx & 63;
    if (tid < DD) {
        float v = (tid < EE) ? row_embed[ri * EE + tid] : col_embed[cj * EE + (tid - EE)];
        cur[tid] = has ? v : 0.0f;
    }
    if (tid < NHD) headden[tid] = 0.0f;
    if (tid < DD)  headnum[tid >> 4][tid & 15] = 0.0f;
    __syncthreads();

    // ---- q = cur @ Wq + bq
    if (tid < DD) {
        float s = bq[tid];
        for (int d = 0; d < DD; ++d) s += cur[d] * Wq[d * DD + tid];
        q[tid] = s;
    }
    __syncthreads();

    // ---- attention over 1024 unique (color,row) keys, weighted by counts
    const int h = tid >> 6;        // head 0..3
    const int l = tid & 63;        // 64 lanes per head, 16 table entries each
    const int* cin = hist_in + b * T;

    float sc[16];
    float m = -1e30f;
#pragma unroll
    for (int it = 0; it < 16; ++it) {
        const int t = l + (it << 6);
        float s = 0.0f;
        for (int d = 0; d < HDD; ++d) s += q[h * HDD + d] * Ktab[t * DD + h * HDD + d];
        s *= 0.25f;                // 1/sqrt(HD)
        sc[it] = s;
        m = fmaxf(m, s);
    }
    red[tid] = m;
    __syncthreads();
    for (int off = 32; off >= 1; off >>= 1) {
        if (l < off) red[tid] = fmaxf(red[tid], red[tid + off]);
        __syncthreads();
    }
    if (l == 0) headmax[h] = red[h << 6];
    __syncthreads();
    m = headmax[h];

    float den = 0.0f;
    float num[HDD];
#pragma unroll
    for (int d = 0; d < HDD; ++d) num[d] = 0.0f;
#pragma unroll
    for (int it = 0; it < 16; ++it) {
        const int t = l + (it << 6);
        const float c = (float)cin[t];
        if (c != 0.0f) {
            const float w = c * __expf(sc[it] - m);
            den += w;
            for (int d = 0; d < HDD; ++d) num[d] += w * Vtab[t * DD + h * HDD + d];
        }
    }
    atomicAdd(&headden[h], den);
#pragma unroll
    for (int d = 0; d < HDD; ++d) atomicAdd(&headnum[h][d], num[d]);

    // ---- output_context = (1/4096) * hist_out . Fout   (4 chunks x 64 cols)
    {
        const int* cout = hist_out + b * T;
        const int col = tid & 63, chunk = tid >> 6;
        float s = 0.0f;
        for (int t = chunk * 256; t < chunk * 256 + 256; ++t) {
            const float c = (float)cout[t];
            if (c != 0.0f) s += c * Fout[t * DD + col];
        }
        ctxpart[chunk][col] = s;
    }
    __syncthreads();

    if (tid < DD) {
        attended[tid] = headnum[tid >> 4][tid & 15] / headden[tid >> 4];
        ctx[tid] = (ctxpart[0][tid] + ctxpart[1][tid] + ctxpart[2][tid] + ctxpart[3][tid])
                   * (1.0f / 4096.0f);
    }
    __syncthreads();

    // ---- att_o = attended @ Wo + bo
    if (tid < DD) {
        float s = bo[tid];
        for (int d = 0; d < DD; ++d) s += attended[d] * Wo[d * DD + tid];
        att_o[tid] = s;
    }
    __syncthreads();

    if (tid < DD)            combined[tid] = att_o[tid];
    else if (tid < 2 * DD)   combined[tid] = ctx[tid - DD];
    else if (tid < 3 * DD)   combined[tid] = cur[tid - 2 * DD];
    __syncthreads();

    // ---- head MLPs: hidden layers (256 wide, one neuron per thread)
    {
        float sp = p_b1[tid], sv = v_b1[tid];
        for (int d = 0; d < 3 * DD; ++d) {
            const float c = combined[d];
            sp += c * p_W1[d * HID + tid];
            sv += c * v_W1[d * HID + tid];
        }
        h1p[tid] = fmaxf(sp, 0.0f);
        h1v[tid] = fmaxf(sv, 0.0f);
    }
    __syncthreads();

    // ---- logits
    if (tid < NC) {
        float s = p_b2[tid];
        for (int d = 0; d < HID; ++d) s += h1p[d] * p_W2[d * NC + tid];
        out[b * (NC + 1) + tid] = s;
    }
    // ---- value (tree reduce)
    red[tid] = h1v[tid] * v_W2[tid];
    __syncthreads();
    for (int off = 128; off >= 1; off >>= 1) {
        if (tid < off) red[tid] += red[tid + off];
        __syncthreads();
    }
    if (tid == 0) out[b * (NC + 1) + NC] = red[0] + v_b2[0];
}

// ---------------------------------------------------------------------------
// Host-side launch
// ---------------------------------------------------------------------------
extern "C" void kernel_launch(void* const* d_in, const int* in_sizes, int n_in,
                              void* d_out, int out_size, void* d_ws, size_t ws_size,
                              hipStream_t stream) {
    const int*   obs         = (const int*)  d_in[0];
    const float* color_embed = (const float*)d_in[1];
    const float* row_embed   = (const float*)d_in[2];
    const float* col_embed   = (const float*)d_in[3];
    const float* in_W1 = (const float*)d_in[4];
    const float* in_b1 = (const float*)d_in[5];
    const float* in_W2 = (const float*)d_in[6];
    const float* in_b2 = (const float*)d_in[7];
    const float* Wq = (const float*)d_in[8];
    const float* bq = (const float*)d_in[9];
    const float* Wk = (const float*)d_in[10];
    const float* bk = (const float*)d_in[11];
    const float* Wv = (const float*)d_in[12];
    const float* bv = (const float*)d_in[13];
    const float* Wo = (const float*)d_in[14];
    const float* bo = (const float*)d_in[15];
    const float* out_W1 = (const float*)d_in[16];
    const float* out_b1 = (const float*)d_in[17];
    const float* out_W2 = (const float*)d_in[18];
    const float* out_b2 = (const float*)d_in[19];
    const float* p_W1 = (const float*)d_in[20];
    const float* p_b1 = (const float*)d_in[21];
    const float* p_W2 = (const float*)d_in[22];
    const float* p_b2 = (const float*)d_in[23];
    const float* v_W1 = (const float*)d_in[24];
    const float* v_b1 = (const float*)d_in[25];
    const float* v_W2 = (const float*)d_in[26];
    const float* v_b2 = (const float*)d_in[27];
    float* out = (float*)d_out;

    // Workspace layout (float offsets)
    float* ws    = (float*)d_ws;
    float* X     = ws;                 // 1024 x 64
    float* H1in  = ws + 65536;         // 1024 x 256
    float* Fin   = ws + 327680;        // 1024 x 64
    float* Ktab  = ws + 393216;        // 1024 x 64
    float* Vtab  = ws + 458752;        // 1024 x 64
    float* H1out = ws + 524288;        // 1024 x 256
    float* Fout  = ws + 786432;        // 1024 x 64
    int*   hist_in  = (int*)(ws + 851968);   // 128 x 1024
    int*   hist_out = (int*)(ws + 983040);   // 128 x 1024
    int*   fidx     = (int*)(ws + 1114112);  // 128

    // 1) X table
    build_x_kernel<<<256, 256, 0, stream>>>(color_embed, row_embed, X);

    // 2) Table GEMMs via fp32 WMMA (8 waves/block -> 8 tiles/block)
    // H1in = relu(X @ in_W1 + in_b1)        M=1024 K=64  N=256 -> 1024 tiles
    wmma_gemm_f32_kernel<<<128, 256, 0, stream>>>(X, in_W1, in_b1, H1in, 1024, 64, 256, 1);
    // Fin = H1in @ in_W2 + in_b2            M=1024 K=256 N=64  -> 256 tiles
    wmma_gemm_f32_kernel<<<32, 256, 0, stream>>>(H1in, in_W2, in_b2, Fin, 1024, 256, 64, 0);
    // Ktab = Fin @ Wk + bk ; Vtab = Fin @ Wv + bv
    wmma_gemm_f32_kernel<<<32, 256, 0, stream>>>(Fin, Wk, bk, Ktab, 1024, 64, 64, 0);
    wmma_gemm_f32_kernel<<<32, 256, 0, stream>>>(Fin, Wv, bv, Vtab, 1024, 64, 64, 0);
    // H1out = relu(X @ out_W1 + out_b1) ; Fout = H1out @ out_W2 + out_b2
    wmma_gemm_f32_kernel<<<128, 256, 0, stream>>>(X, out_W1, out_b1, H1out, 1024, 64, 256, 1);
    wmma_gemm_f32_kernel<<<32, 256, 0, stream>>>(H1out, out_W2, out_b2, Fout, 1024, 256, 64, 0);

    // 3) Histograms + first-mask index (one pass over obs: the only big read)
    hist_kernel<<<BB, 256, 0, stream>>>(obs, hist_in, hist_out, fidx);

    // 4) Per-batch attention + heads
    finalize_kernel<<<BB, 256, 0, stream>>>(row_embed, col_embed, Wq, bq, Wo, bo,
                                            Ktab, Vtab, Fout,
                                            hist_in, hist_out, fidx,
                                            p_W1, p_b1, p_W2, p_b2,
                                            v_W1, v_b1, v_W2, v_b2, out);
}